// GATRNNWithAttentionModel_13864154432151
// MI455X (gfx1250) — compile-verified
//
#include <hip/hip_runtime.h>
#include <hip/hip_bf16.h>

// ---------------- problem constants (from reference) ----------------
constexpr int kN   = 100000;   // nodes
constexpr int kE   = 1600000;  // edges
constexpr int kIC  = 128;      // IN_CH
constexpr int kHD  = 128;      // HEADS*HID
constexpr int kG   = 384;      // live gate columns (i, c_hat, o)
constexpr int kOUT = 16;       // OUT_CH

typedef __attribute__((ext_vector_type(2))) float v2f;
typedef __attribute__((ext_vector_type(8))) float v8f;

// f32 WMMA: D(16x16,f32) = A(16x4,f32) * B(4x16,f32) + C
// 8-arg form: (neg_a, A, neg_b, B, c_mod, C, reuse_a, reuse_b)
__device__ __forceinline__ v8f wmma_f32(v2f a, v2f b, v8f c) {
  return __builtin_amdgcn_wmma_f32_16x16x4_f32(
      false, a, false, b, (short)0, c, false, false);
}

// monotonic unsigned encoding of float for atomicMax-based segment max
__device__ __forceinline__ unsigned fenc(float f) {
  unsigned u = __float_as_uint(f);
  return (u & 0x80000000u) ? ~u : (u | 0x80000000u);
}
__device__ __forceinline__ float fdec(unsigned k) {
  return __uint_as_float((k & 0x80000000u) ? (k ^ 0x80000000u) : ~k);
}
__device__ __forceinline__ float sigmoidf(float x) {
  return 1.0f / (1.0f + __expf(-x));
}

// ---------------- K0: init accumulators ----------------
__global__ __launch_bounds__(256) void k_init(float* __restrict__ gat,
                                              unsigned* __restrict__ maxe,
                                              float* __restrict__ denom) {
  size_t i = (size_t)blockIdx.x * blockDim.x + threadIdx.x;
  if (i < (size_t)kN * kHD) gat[i] = 0.0f;
  if (i < (size_t)kN * 2) { maxe[i] = 0x007FFFFFu; denom[i] = 0.0f; }
}

// ---------------- K1: h = x @ W  (+ a_src, a_dst) ----------------
// block = 256 threads (8 waves) per 16-node tile; wave w -> column tile w*16
__global__ __launch_bounds__(256) void k_node_transform(
    const float* __restrict__ x, const float* __restrict__ W,
    const float* __restrict__ att_src, const float* __restrict__ att_dst,
    float* __restrict__ h, float* __restrict__ asrc, float* __restrict__ adst) {
  __shared__ float xs[16][132];      // padded: stride 132 -> no LDS bank conflicts
  __shared__ float as_s[16][2];
  __shared__ float ad_s[16][2];
  const int tid = threadIdx.x;
  const int node0 = blockIdx.x * 16;

  if (tid < 32) { as_s[tid >> 1][tid & 1] = 0.0f; ad_s[tid >> 1][tid & 1] = 0.0f; }
  for (int i = tid; i < 16 * kIC; i += 256) {
    int r = i >> 7, c = i & 127;
    xs[r][c] = x[(size_t)(node0 + r) * kIC + c];
  }
  __syncthreads();

  const int wave = tid >> 5, lane = tid & 31;
  const int n = lane & 15, kh = lane >> 4, kb = 2 * kh;
  const int c0 = wave * 16;

  v8f acc = {};
  for (int kk = 0; kk < kIC; kk += 4) {
    v2f a = { xs[n][kk + kb], xs[n][kk + kb + 1] };
    v2f b = { W[(size_t)(kk + kb) * kHD + c0 + n],
              W[(size_t)(kk + kb + 1) * kHD + c0 + n] };
    acc = wmma_f32(a, b, acc);
  }

  const int head = c0 >> 6;
  const float asv = att_src[head * 64 + (c0 & 63) + n];
  const float adv = att_dst[head * 64 + (c0 & 63) + n];
#pragma unroll
  for (int r = 0; r < 8; r++) {
    int m = r + 8 * kh;                    // C/D layout: lanes16-31 hold M+8
    h[(size_t)(node0 + m) * kHD + c0 + n] = acc[r];
    atomicAdd(&as_s[m][head], acc[r] * asv);
    atomicAdd(&ad_s[m][head], acc[r] * adv);
  }
  __syncthreads();
  if (tid < 32) {
    asrc[(size_t)(node0 + (tid >> 1)) * 2 + (tid & 1)] = as_s[tid >> 1][tid & 1];
    adst[(size_t)(node0 + (tid >> 1)) * 2 + (tid & 1)] = ad_s[tid >> 1][tid & 1];
  }
}

// ---------------- K2: per-edge leaky-relu score + segment max ----------------
__global__ __launch_bounds__(256) void k_edge_alpha(
    const int* __restrict__ ei, const float* __restrict__ asrc,
    const float* __restrict__ adst, float* __restrict__ alpha,
    unsigned* __restrict__ maxe) {
  int e = blockIdx.x * blockDim.x + threadIdx.x;
  if (e >= kE + kN) return;
  int s = (e < kE) ? ei[e] : (e - kE);        // self-loops appended
  int d = (e < kE) ? ei[kE + e] : (e - kE);
#pragma unroll
  for (int hh = 0; hh < 2; hh++) {
    float raw = asrc[(size_t)s * 2 + hh] + adst[(size_t)d * 2 + hh];
    raw = (raw > 0.0f) ? raw : 0.2f * raw;    // leaky_relu 0.2
    alpha[(size_t)e * 2 + hh] = raw;
    atomicMax(&maxe[(size_t)d * 2 + hh], fenc(raw));
  }
}

// ---------------- K3: exp(alpha - max) + segment sum ----------------
__global__ __launch_bounds__(256) void k_edge_exp(
    const int* __restrict__ ei, float* __restrict__ alpha,
    const unsigned* __restrict__ maxe, float* __restrict__ denom) {
  int e = blockIdx.x * blockDim.x + threadIdx.x;
  if (e >= kE + kN) return;
  int d = (e < kE) ? ei[kE + e] : (e - kE);
#pragma unroll
  for (int hh = 0; hh < 2; hh++) {
    float m = fdec(maxe[(size_t)d * 2 + hh]);
    float a = __expf(alpha[(size_t)e * 2 + hh] - m);
    alpha[(size_t)e * 2 + hh] = a;
    atomicAdd(&denom[(size_t)d * 2 + hh], a);
  }
}

// ---------------- K4: gat[dst] += w * h[src] ----------------
// 4 threads per edge, 32 contiguous channels each (float4 gathers)
__global__ __launch_bounds__(256) void k_aggregate(
    const int* __restrict__ ei, const float* __restrict__ h,
    const float* __restrict__ alpha, const float* __restrict__ denom,
    float* __restrict__ gat) {
  long long t = (long long)blockIdx.x * blockDim.x + threadIdx.x;
  if (t >= (long long)(kE + kN) * 4) return;
  int e = (int)(t >> 2), q = (int)(t & 3);
  int s = (e < kE) ? ei[e] : (e - kE);
  int d = (e < kE) ? ei[kE + e] : (e - kE);
  int head = q >> 1;
  float w = alpha[(size_t)e * 2 + head] /
            (denom[(size_t)d * 2 + head] + 1e-16f);
  const float4* hp = (const float4*)(h + (size_t)s * kHD + q * 32);
  float* gp = gat + (size_t)d * kHD + q * 32;
#pragma unroll
  for (int j = 0; j < 8; j++) {
    float4 v = hp[j];
    atomicAdd(gp + 4 * j + 0, w * v.x);
    atomicAdd(gp + 4 * j + 1, w * v.y);
    atomicAdd(gp + 4 * j + 2, w * v.z);
    atomicAdd(gp + 4 * j + 3, w * v.w);
  }
}

// ---------------- K5: transpose live gate blocks of W_ih ----------------
// Wt[k][c] = W_ih[row(c)][k], c in [0,384): i=rows 0..127, c_hat=256..383, o=384..511
__global__ __launch_bounds__(256) void k_transpose_wih(
    const float* __restrict__ W_ih, float* __restrict__ Wt) {
  int i = blockIdx.x * blockDim.x + threadIdx.x;
  if (i >= kHD * kG) return;
  int c = i % kG, k = i / kG;
  int row = c + (c >= 128 ? 128 : 0);
  Wt[(size_t)k * kG + c] = W_ih[(size_t)row * kHD + k];
}

// ---------------- K6: fused relu + gates GEMM + LSTM + FC ----------------
// block = 8 waves per 16-node tile; waves cover 24 gate column tiles, then
// wave 0 runs the 128->16 FC WMMA chain. (f gate / W_hh / v_attn are dead.)
__global__ __launch_bounds__(256) void k_lstm_fc(
    const float* __restrict__ gat, const float* __restrict__ gat_bias,
    const float* __restrict__ Wt, const float* __restrict__ b_ih,
    const float* __restrict__ b_hh, const float* __restrict__ W_fc,
    const float* __restrict__ b_fc, float* __restrict__ out) {
  __shared__ float gs[16][132];
  __shared__ float gates_s[16][388];   // stride 388: lanes 0-15 / 16-31 bank-disjoint
  __shared__ float hs[16][132];
  const int tid = threadIdx.x;
  const int node0 = blockIdx.x * 16;

  for (int i = tid; i < 16 * kHD; i += 256) {
    int r = i >> 7, c = i & 127;
    float v = gat[(size_t)(node0 + r) * kHD + c] + gat_bias[c];
    gs[r][c] = (v > 0.0f) ? v : 0.0f;         // relu after GAT
  }
  __syncthreads();

  const int wave = tid >> 5, lane = tid & 31;
  const int n = lane & 15, kh = lane >> 4, kb = 2 * kh;

  for (int t = wave; t < 24; t += 8) {
    const int gc0 = t * 16;
    v8f acc = {};
    for (int kk = 0; kk < kHD; kk += 4) {
      v2f a = { gs[n][kk + kb], gs[n][kk + kb + 1] };
      v2f b = { Wt[(size_t)(kk + kb) * kG + gc0 + n],
                Wt[(size_t)(kk + kb + 1) * kG + gc0 + n] };
      acc = wmma_f32(a, b, acc);
    }
    const int gc = gc0 + n;
    const int row = gc + (gc >= 128 ? 128 : 0);
    const float bias = b_ih[row] + b_hh[row];
#pragma unroll
    for (int r = 0; r < 8; r++) gates_s[r + 8 * kh][gc] = acc[r] + bias;
  }
  __syncthreads();

  for (int i = tid; i < 16 * kHD; i += 256) {
    int m = i >> 7, d = i & 127;
    float ig = gates_s[m][d];
    float ch = gates_s[m][128 + d];
    float og = gates_s[m][256 + d];
    float c  = sigmoidf(ig) * tanhf(ch);
    hs[m][d] = sigmoidf(og) * tanhf(c);       // seq len 1 -> context == hn
  }
  __syncthreads();

  if (wave == 0) {   // wave-uniform branch: EXEC stays all-ones for WMMA
    v8f acc = {};
    for (int kk = 0; kk < kHD; kk += 4) {
      v2f a = { hs[n][kk + kb], hs[n][kk + kb + 1] };
      v2f b = { W_fc[(size_t)n * kHD + kk + kb],
                W_fc[(size_t)n * kHD + kk + kb + 1] };
      acc = wmma_f32(a, b, acc);
    }
    const float bb = b_fc[n];
#pragma unroll
    for (int r = 0; r < 8; r++)
      out[(size_t)(node0 + r + 8 * kh) * kOUT + n] = acc[r] + bb;
  }
}

// ---------------- host launcher ----------------
extern "C" void kernel_launch(void* const* d_in, const int* in_sizes, int n_in,
                              void* d_out, int out_size, void* d_ws, size_t ws_size,
                              hipStream_t stream) {
  const float* x        = (const float*)d_in[0];
  const int*   ei       = (const int*)d_in[1];   // (2,E): row0=src, row1=dst
  const float* W        = (const float*)d_in[2];
  const float* att_src  = (const float*)d_in[3];
  const float* att_dst  = (const float*)d_in[4];
  const float* gat_bias = (const float*)d_in[5];
  const float* W_ih     = (const float*)d_in[6];
  // d_in[7] = W_hh (dead: zero initial hidden state)
  const float* b_ih     = (const float*)d_in[8];
  const float* b_hh     = (const float*)d_in[9];
  // d_in[10] = v_attn (dead: softmax over sequence length 1)
  const float* W_fc     = (const float*)d_in[11];
  const float* b_fc     = (const float*)d_in[12];
  float* out = (float*)d_out;

  // workspace carve-up (~119 MB total)
  float*    h     = (float*)d_ws;                          // N*128
  float*    asrc  = h + (size_t)kN * kHD;                  // N*2
  float*    adst  = asrc + (size_t)kN * 2;                 // N*2
  unsigned* maxe  = (unsigned*)(adst + (size_t)kN * 2);    // N*2
  float*    denom = (float*)(maxe + (size_t)kN * 2);       // N*2
  float*    alpha = denom + (size_t)kN * 2;                // (E+N)*2
  float*    gat   = alpha + (size_t)(kE + kN) * 2;         // N*128
  float*    Wt    = gat + (size_t)kN * kHD;                // 128*384

  k_init<<<(kN * kHD + 255) / 256, 256, 0, stream>>>(gat, maxe, denom);
  k_node_transform<<<kN / 16, 256, 0, stream>>>(x, W, att_src, att_dst, h, asrc, adst);
  k_transpose_wih<<<(kHD * kG + 255) / 256, 256, 0, stream>>>(W_ih, Wt);
  k_edge_alpha<<<(kE + kN + 255) / 256, 256, 0, stream>>>(ei, asrc, adst, alpha, maxe);
  k_edge_exp<<<(kE + kN + 255) / 256, 256, 0, stream>>>(ei, alpha, maxe, denom);
  long long tot = (long long)(kE + kN) * 4;
  k_aggregate<<<(unsigned)((tot + 255) / 256), 256, 0, stream>>>(ei, h, alpha, denom, gat);
  k_lstm_fc<<<kN / 16, 256, 0, stream>>>(gat, gat_bias, Wt, b_ih, b_hh, W_fc, b_fc, out);
}